// MLPEncoder_27376121544732
// MI455X (gfx1250) — compile-verified
//
#include <hip/hip_runtime.h>
#include <hip/hip_bf16.h>
#include <stdint.h>

// Problem constants (from reference): B=2048, L=128, V=30522, D=768
#define BB   2048
#define LL   128
#define DD   768
#define ROWB (DD * 4)   // 3072 bytes per embedding row
#define NBUF 8          // LDS row buffers (24 KB)
#define PF   6          // rows in flight per wave (NBUF-PF = 2 slots of slack)
#define NT   192        // 6 wave32s; 192 lanes * 16B = one full row per issue

__global__ __launch_bounds__(NT) void embed_maskmean_kernel(
    const int*   __restrict__ ids,    // [B, L]
    const int*   __restrict__ mask,   // [B, L], left-contiguous 1s
    const float* __restrict__ emb,    // [V, D]
    float*       __restrict__ out)    // [B, D]
{
    __shared__ __align__(16) float s_buf[NBUF][DD];
    __shared__ int s_ids[LL];
    __shared__ int s_len;

    const int b   = blockIdx.x;
    const int tid = threadIdx.x;

    if (tid == 0) s_len = 0;
    __syncthreads();
    if (tid < LL) {
        s_ids[tid] = ids[b * LL + tid];
        if (mask[b * LL + tid] != 0) atomicAdd(&s_len, 1);   // ds_add_u32
    }
    __syncthreads();                 // last barrier in the kernel
    const int len = s_len;           // uniform; in [1, L]

    // Wave-local partition: wave w owns row bytes [512w, 512w+512).
    // Each lane copies (async) and later reads back the SAME 16B chunk,
    // so no cross-wave synchronization is needed in the pipeline.
    const int      chunkByte = tid * 16;
    const unsigned ldsChunk  = (unsigned)(uintptr_t)(&s_buf[0][0]) + (unsigned)chunkByte;
    const uint64_t embBase   = (uint64_t)(uintptr_t)emb;

    auto issue_row = [&](int row) {
        const uint64_t g = embBase
                         + (uint64_t)(unsigned)s_ids[row] * (uint64_t)ROWB
                         + (uint64_t)chunkByte;
        const unsigned d = ldsChunk + (unsigned)((row & (NBUF - 1)) * ROWB);
        asm volatile("global_load_async_to_lds_b128 %0, %1, off"
                     :
                     : "v"(d), "v"(g)
                     : "memory");   // tracked on ASYNCcnt; one instr per wave per row
    };

    // Fill the pipeline: up to PF rows in flight per wave.
    const int npf = (len < PF) ? len : PF;
    for (int i = 0; i < npf; ++i) issue_row(i);

    float4 acc = make_float4(0.f, 0.f, 0.f, 0.f);

    int l = 0;
    const int lmain = (len > PF) ? (len - PF) : 0;

    // Steady state: row l is the oldest of exactly PF outstanding copies.
    for (; l < lmain; ++l) {
        asm volatile("s_wait_asynccnt 0x5" ::: "memory");   // oldest (row l) landed
        const float4 v = *(const float4*)&s_buf[l & (NBUF - 1)][tid * 4];
        acc.x += v.x; acc.y += v.y; acc.z += v.z; acc.w += v.w;
        // Overwrites slot (l-2)&7: its ds_load was consumed (dscnt-waited)
        // two iterations ago, before this asm in program order.
        issue_row(l + PF);
    }

    // Drain: fewer than PF rows remain in flight.
    for (; l < len; ++l) {
        const int allow = len - l - 1;   // uniform, 0..PF-1
        if      (allow >= 5) asm volatile("s_wait_asynccnt 0x5" ::: "memory");
        else if (allow == 4) asm volatile("s_wait_asynccnt 0x4" ::: "memory");
        else if (allow == 3) asm volatile("s_wait_asynccnt 0x3" ::: "memory");
        else if (allow == 2) asm volatile("s_wait_asynccnt 0x2" ::: "memory");
        else if (allow == 1) asm volatile("s_wait_asynccnt 0x1" ::: "memory");
        else                 asm volatile("s_wait_asynccnt 0x0" ::: "memory");
        const float4 v = *(const float4*)&s_buf[l & (NBUF - 1)][tid * 4];
        acc.x += v.x; acc.y += v.y; acc.z += v.z; acc.w += v.w;
    }

    const float inv = 1.0f / (float)(len > 0 ? len : 1);
    float4 r;
    r.x = acc.x * inv; r.y = acc.y * inv; r.z = acc.z * inv; r.w = acc.w * inv;
    *(float4*)&out[b * DD + tid * 4] = r;   // coalesced global_store_b128
}

extern "C" void kernel_launch(void* const* d_in, const int* in_sizes, int n_in,
                              void* d_out, int out_size, void* d_ws, size_t ws_size,
                              hipStream_t stream) {
    (void)in_sizes; (void)n_in; (void)d_ws; (void)ws_size; (void)out_size;
    const int*   ids  = (const int*)d_in[0];    // tag_input_ids      [B, L]
    const int*   mask = (const int*)d_in[1];    // tag_attention_mask [B, L]
    const float* emb  = (const float*)d_in[2];  // emb_weight         [V, D]
    float*       out  = (float*)d_out;          // [B, D] fp32

    dim3 grid(BB), block(NT);
    embed_maskmean_kernel<<<grid, block, 0, stream>>>(ids, mask, emb, out);
}